// SNAMD_39273180954722
// MI455X (gfx1250) — compile-verified
//
#include <hip/hip_runtime.h>

typedef __attribute__((ext_vector_type(2))) float v2f;
typedef __attribute__((ext_vector_type(4))) float v4f;
typedef __attribute__((ext_vector_type(8))) float v8f;

#define C_DIM 768
#define H_DIM 32
#define NPIX  1024
#define NTOK  1025

__global__ __launch_bounds__(128) void snamd_embed_wmma(
    const float* __restrict__ feat0,
    const float* __restrict__ feat1,
    float* __restrict__ out)
{
    __shared__ int   s_off[64];        // element offset of nb pixel row into fb, or -1 (OOB)
    __shared__ float s_norm[64];       // per-neighbor squared norm over C
    __shared__ float s_G[16][64];      // Gram: centers x neighbors
    __shared__ float s_W[16][64];      // exp weights
    __shared__ float s_den[16][3];     // denominators for r=1,3,5
    __shared__ float s_M[16][66];      // combined coefficients (row-padded)

    const int tid   = threadIdx.x;
    const int wave  = tid >> 5;
    const int lane  = tid & 31;
    const int l16   = lane & 15;
    const int khalf = (lane >> 4) << 1;     // 0 for lanes 0-15, 2 for lanes 16-31

    const int id    = blockIdx.x;
    const int layer = id >> 8;              // 256 blocks per layer
    const int batch = (id >> 6) & 3;
    const int blk   = id & 63;
    const int bh0   = (blk >> 3) << 2;
    const int bw0   = (blk & 7) << 2;

    const float* fb = (layer == 0 ? feat0 : feat1) + (size_t)batch * NTOK * C_DIM;

    // ---- neighborhood offsets (8x8 block around the 4x4 center tile) ----
    if (tid < 64) {
        const int nr = tid >> 3, nc = tid & 7;
        const int h = bh0 - 2 + nr, w = bw0 - 2 + nc;
        const bool ok = (h >= 0) & (h < H_DIM) & (w >= 0) & (w < H_DIM);
        s_off[tid] = ok ? (1 + h * H_DIM + w) * C_DIM : -1;
    }
    __syncthreads();

    // ---- per-pixel squared norms (OOB pixels are zero vectors) ----
    if (tid < 64) {
        const int off = s_off[tid];
        float acc = 0.f;
        if (off >= 0) {
            const v4f* p = (const v4f*)(fb + off);
            for (int i = 0; i < C_DIM / 4; ++i) {
                v4f v = p[i];
                acc += v.x * v.x + v.y * v.y + v.z * v.z + v.w * v.w;
            }
        }
        s_norm[tid] = acc;
    }

    // ---- phase 1: G = F_centers (16xC) * F_nbr^T (Cx64), wave w -> cols [16w,16w+16) ----
    {
        const int mh = bh0 + (l16 >> 2), mw = bw0 + (l16 & 3);
        const float* arow = fb + (size_t)(1 + mh * H_DIM + mw) * C_DIM;
        const int   n     = (wave << 4) + l16;
        const int   boff  = s_off[n];
        const bool  bok   = (boff >= 0);
        const float* brow = fb + (bok ? boff : 0);   // safe address; masked below

        v8f acc = {};
        #pragma unroll 4
        for (int kk = 0; kk < C_DIM; kk += 4) {
            v2f a = *(const v2f*)(arow + kk + khalf);
            v2f braw = *(const v2f*)(brow + kk + khalf);
            v2f b;
            b.x = bok ? braw.x : 0.f;
            b.y = bok ? braw.y : 0.f;
            acc = __builtin_amdgcn_wmma_f32_16x16x4_f32(
                false, a, false, b, (short)0, acc, false, false);
        }
        const int grow = (lane >> 4) << 3;      // 0 or 8
        #pragma unroll
        for (int i = 0; i < 8; ++i)
            s_G[grow + i][(wave << 4) + l16] = acc[i];
    }
    __syncthreads();

    // ---- phase 2a: weights + denominators (one thread per center) ----
    if (tid < 16) {
        const int m  = tid;
        const int mr = m >> 2, mc = m & 3;
        const float nm = s_norm[(mr + 2) * 8 + (mc + 2)];
        float d1 = 0.f, d3 = 0.f, d5 = 0.f;
        for (int n = 0; n < 64; ++n) {
            const int nr = n >> 3, nc = n & 7;
            int dr = nr - mr - 2, dc = nc - mc - 2;
            const int adr = dr < 0 ? -dr : dr;
            const int adc = dc < 0 ? -dc : dc;
            float ss = nm + s_norm[n] - 2.f * s_G[m][n];
            ss = fmaxf(ss, 0.f);
            const float w = __expf(-sqrtf(ss));
            s_W[m][n] = w;
            if (adr <= 2 && adc <= 2) {
                d5 += w;
                if (adr <= 1 && adc <= 1) {
                    d3 += w;
                    if (adr == 0 && adc == 0) d1 += w;
                }
            }
        }
        s_den[m][0] = d1; s_den[m][1] = d3; s_den[m][2] = d5;
    }
    __syncthreads();

    // ---- phase 2b: combined coefficient matrix M[m][n] ----
    for (int p = tid; p < 1024; p += 128) {
        const int m = p >> 6, n = p & 63;
        const int mr = m >> 2, mc = m & 3;
        const int nr = n >> 3, nc = n & 7;
        int dr = nr - mr - 2, dc = nc - mc - 2;
        const int adr = dr < 0 ? -dr : dr;
        const int adc = dc < 0 ? -dc : dc;
        float coef = 0.f;
        if (adr <= 2 && adc <= 2) {
            coef += 1.f / (3.f * s_den[m][2]);
            if (adr <= 1 && adc <= 1) {
                coef += 1.f / (3.f * s_den[m][1]);
                if (adr == 0 && adc == 0) coef += 1.f / (3.f * s_den[m][0]);
            }
        }
        s_M[m][n] = s_W[m][n] * coef;
    }
    __syncthreads();

    // ---- phase 3: Out (16xC) = M (16x64) * F_nbr (64xC); 12 channel tiles per wave ----
    for (int t = 0; t < 12; ++t) {
        const int ch0 = ((wave * 12) + t) << 4;
        v8f acc = {};
        #pragma unroll
        for (int kk = 0; kk < 64; kk += 4) {
            v2f a = *(const v2f*)(&s_M[l16][kk + khalf]);
            const int n0 = kk + khalf;
            const int o0 = s_off[n0], o1 = s_off[n0 + 1];
            const float r0 = fb[(o0 >= 0 ? o0 : 0) + ch0 + l16];
            const float r1 = fb[(o1 >= 0 ? o1 : 0) + ch0 + l16];
            v2f b;
            b.x = (o0 >= 0) ? r0 : 0.f;
            b.y = (o1 >= 0) ? r1 : 0.f;
            acc = __builtin_amdgcn_wmma_f32_16x16x4_f32(
                false, a, false, b, (short)0, acc, false, false);
        }
        const int grow = (lane >> 4) << 3;
        #pragma unroll
        for (int i = 0; i < 8; ++i) {
            const int m  = grow + i;
            const int hw = (bh0 + (m >> 2)) * H_DIM + (bw0 + (m & 3));
            const size_t oidx =
                ((((size_t)batch * NPIX + hw) * 2) + layer) * C_DIM + ch0 + l16;
            out[oidx] = acc[i];
        }
    }
}

extern "C" void kernel_launch(void* const* d_in, const int* in_sizes, int n_in,
                              void* d_out, int out_size, void* d_ws, size_t ws_size,
                              hipStream_t stream) {
    (void)in_sizes; (void)n_in; (void)out_size; (void)d_ws; (void)ws_size;
    const float* feat0 = (const float*)d_in[0];
    const float* feat1 = (const float*)d_in[1];
    float* out = (float*)d_out;
    // 2 layers * 4 batches * 64 (4x4) tiles = 512 workgroups, 4 waves each
    snamd_embed_wmma<<<512, 128, 0, stream>>>(feat0, feat1, out);
}